// DeepFilter_16587163697924
// MI455X (gfx1250) — compile-verified
//
#include <hip/hip_runtime.h>

// DeepFilter on MI455X (gfx1250).
// Identity: both operands are shifted identically per tap, so the output is a
// 3x5 zero-padded box sum of the elementwise maps Q_r = ir*fr - ii*fi, Q_i = 2*ir*fi.
// Memory-bound (~0.44 GB traffic -> ~19us @ 23.3 TB/s). Strategy:
//   1) stage Q_r/Q_i tile (+halo) in LDS; interior via b128 loads (4x fewer VMEM ops),
//   2) 3-tap F-direction sum in LDS (cheap VALU),
//   3) 5-tap T-direction sum as a banded matmul via V_WMMA_F32_16X16X4_F32,
//   4) coalesced fp32 stores straight from the WMMA C/D register layout.

typedef float v2f __attribute__((ext_vector_type(2)));
typedef float v4f __attribute__((ext_vector_type(4)));
typedef float v8f __attribute__((ext_vector_type(8)));

#define NB   16
#define NF   257
#define NT   4000
#define FT   16     // F rows per block
#define TT   128    // T cols per block (8 waves x 16)
#define QROWS 18    // FT + 2 halo
#define QCOLS 132   // TT + 4 halo
#define LSTR  133   // padded LDS row stride (odd: conflict-free row-strided reads)
#define PROWS 16

__global__ __launch_bounds__(256)
void deepfilter_gfx1250(const float* __restrict__ ir,
                        const float* __restrict__ ii,
                        const float* __restrict__ fr,
                        const float* __restrict__ fi,
                        float* __restrict__ out)
{
    __shared__ float QR[QROWS * LSTR];
    __shared__ float QI[QROWS * LSTR];
    __shared__ float PR[PROWS * LSTR];
    __shared__ float PI[PROWS * LSTR];

    const int tid   = threadIdx.x;
    const int tBlk  = blockIdx.x;          // 0..31
    const int fBlk  = blockIdx.y;          // 0..16
    const int b     = blockIdx.z;          // 0..15
    const int fbase = fBlk * FT;
    const int tbase = tBlk * TT;
    const int ibase = b * (NF * NT);

    // ---- Phase 1a: interior via 16B-aligned b128 loads ---------------------
    // 18 rows x 32 float4 chunks; gt%4==0 and NT%4==0 -> chunk all-in or all-out.
    for (int idx = tid; idx < QROWS * (TT / 4); idx += 256) {
        const int r   = idx >> 5;           // 0..17
        const int c4  = idx & 31;           // float4 chunk within row
        const int gf  = fbase - 1 + r;
        const int gt  = tbase + 4 * c4;
        v4f xr = {0.f,0.f,0.f,0.f}, xi = xr, wr = xr, wi = xr;
        if (gf >= 0 && gf < NF && gt < NT) {
            const int e = ibase + gf * NT + gt;      // multiple of 4 -> 16B aligned
            xr = *(const v4f*)(ir + e);
            xi = *(const v4f*)(ii + e);
            wr = *(const v4f*)(fr + e);
            wi = *(const v4f*)(fi + e);
        }
        const int o = r * LSTR + 2 + 4 * c4;         // interior starts at LDS col 2
#pragma unroll
        for (int k = 0; k < 4; ++k) {
            QR[o + k] = xr[k] * wr[k] - xi[k] * wi[k];
            QI[o + k] = 2.0f * xr[k] * wi[k];
        }
    }

    // ---- Phase 1b: 2-column halos (left cols 0..1, right cols 130..131) ----
    for (int idx = tid; idx < QROWS * 4; idx += 256) {
        const int r  = idx >> 2;
        const int h  = idx & 3;
        const int c  = (h < 2) ? h : (130 + (h - 2));
        const int gf = fbase - 1 + r;
        const int gt = tbase - 2 + c;
        float qr = 0.0f, qi = 0.0f;
        if (gf >= 0 && gf < NF && gt >= 0 && gt < NT) {
            const int e  = ibase + gf * NT + gt;
            const float xr = ir[e], xi = ii[e];
            const float wr = fr[e], wi = fi[e];
            qr = xr * wr - xi * wi;
            qi = 2.0f * xr * wi;
        }
        QR[r * LSTR + c] = qr;
        QI[r * LSTR + c] = qi;
    }
    __syncthreads();

    // ---- Phase 2: 3-tap F-direction sum (P row r <-> output row fbase+r) ---
    for (int idx = tid; idx < PROWS * QCOLS; idx += 256) {
        const int r = idx / QCOLS;
        const int c = idx - r * QCOLS;
        const int o = r * LSTR + c;
        PR[o] = QR[o] + QR[o + LSTR] + QR[o + 2 * LSTR];
        PI[o] = QI[o] + QI[o + LSTR] + QI[o + 2 * LSTR];
    }
    __syncthreads();

    // ---- Phase 3: 5-tap T-direction sum as banded WMMA ---------------------
    // S(16x16) = P(16x20) x Band(20x16),  Band[k][n] = (n <= k <= n+4).
    // Five K=4 chunks of V_WMMA_F32_16X16X4_F32 per channel.
    const int w    = tid >> 5;             // wave 0..7 -> 16-col output slab
    const int lane = tid & 31;
    const int hi   = lane >> 4;            // lane half selects K-pair / M-offset
    const int n    = lane & 15;            // output column within slab / A row M

    const float* prow_r = &PR[n * LSTR + 16 * w];
    const float* prow_i = &PI[n * LSTR + 16 * w];

    v8f accR = {0.f, 0.f, 0.f, 0.f, 0.f, 0.f, 0.f, 0.f};
    v8f accI = {0.f, 0.f, 0.f, 0.f, 0.f, 0.f, 0.f, 0.f};

#pragma unroll
    for (int j = 0; j < 5; ++j) {
        // A operand: lane m holds A[m][kb], A[m][kb+1]; upper half-lanes kb=+2.
        const int kb = 4 * j + 2 * hi;
        v2f aR; aR.x = prow_r[kb]; aR.y = prow_r[kb + 1];
        v2f aI; aI.x = prow_i[kb]; aI.y = prow_i[kb + 1];
        // B operand: VGPR0 = Band row (4j+hi), VGPR1 = Band row (4j+2+hi).
        const int k0 = 4 * j + hi;
        const int k1 = k0 + 2;
        v2f bm;
        bm.x = (k0 >= n && k0 <= n + 4) ? 1.0f : 0.0f;
        bm.y = (k1 >= n && k1 <= n + 4) ? 1.0f : 0.0f;
        accR = __builtin_amdgcn_wmma_f32_16x16x4_f32(false, aR, false, bm,
                                                     (short)0, accR, false, false);
        accI = __builtin_amdgcn_wmma_f32_16x16x4_f32(false, aI, false, bm,
                                                     (short)0, accI, false, false);
    }

    // ---- Phase 4: stores straight from C/D layout --------------------------
    // VGPR r, lane l: row M = r + 8*(l>=16), col N = l&15.
    const int gt = tbase + 16 * w + n;
#pragma unroll
    for (int r = 0; r < 8; ++r) {
        const int gf = fbase + r + 8 * hi;
        if (gf < NF && gt < NT) {
            out[(b * 2 * NF + gf) * NT + gt]      = accR[r];   // real channel
            out[(b * 2 * NF + NF + gf) * NT + gt] = accI[r];   // imag channel
        }
    }
}

extern "C" void kernel_launch(void* const* d_in, const int* in_sizes, int n_in,
                              void* d_out, int out_size, void* d_ws, size_t ws_size,
                              hipStream_t stream)
{
    (void)in_sizes; (void)n_in; (void)out_size; (void)d_ws; (void)ws_size;
    const float* ir = (const float*)d_in[0];
    const float* ii = (const float*)d_in[1];
    const float* fr = (const float*)d_in[2];
    const float* fi = (const float*)d_in[3];
    float* out = (float*)d_out;

    dim3 grid((NT + TT - 1) / TT,   // 32
              (NF + FT - 1) / FT,   // 17
              NB);                  // 16
    deepfilter_gfx1250<<<grid, 256, 0, stream>>>(ir, ii, fr, fi, out);
}